// Img2Text_62663572848742
// MI455X (gfx1250) — compile-verified
//
#include <hip/hip_runtime.h>
#include <math.h>

// ---------------------------------------------------------------------------
// Show-Attend-Tell beam search decoder for MI455X (gfx1250, wave32, WMMA).
// B=16 images, K=3 beams, 50 sequential steps. All GEMMs use
// V_WMMA_F32_16X16X4_F32. Per-step GEMMs other than att2 (MT=1) process all 3 M-tiles (48 beam rows)
// in one wave so each weight matrix is streamed from L2 exactly once per step.
// ---------------------------------------------------------------------------

typedef __attribute__((ext_vector_type(2))) float v2f;
typedef __attribute__((ext_vector_type(8))) float v8f;

#define BIMG   16
#define KBEAM  3
#define NROW   (BIMG * KBEAM)   // 48
#define PPIX   196
#define DENC   2048
#define EDIM   512
#define DDIM   512
#define ADIM   512
#define VVOC   9490
#define XDIM   (EDIM + DENC)    // 2560
#define GDIM   (4 * DDIM)       // 2048
#define SLEN   51               // STEPS+1
#define STARTW 9488
#define ENDW   9489
#define NEGINF (-1e9f)

__device__ __forceinline__ float sigm(float x) { return 1.0f / (1.0f + expf(-x)); }

// ---------------------------------------------------------------------------
// WMMA f32 GEMM: C[M,N] (=|+=) A[M,Kd] * B + bias.
//   BT=false: B stored row-major [Kd,N].
//   BT=true : B stored [N,Kd] (torch weight layout) -> computes A @ B^T;
//             each lane walks contiguous K so cachelines are fully consumed.
//   MT      : number of 16-row M-tiles processed by one wave (B fragment is
//             loaded once and reused by MT independent WMMAs -> weight traffic
//             divided by MT).
// One wave per (M-group, N-tile). M must be a multiple of 16*? (M-groups
// exact here). N edge handled with clamped+masked loads so EXEC stays all-ones
// through the WMMA loop.
// ---------------------------------------------------------------------------
template <int MT, bool BT, bool ACC>
__global__ __launch_bounds__(256) void wmma_gemm_kernel(
    const float* __restrict__ A, int lda,
    const float* __restrict__ B, int ldb,
    const float* __restrict__ bias,
    float* __restrict__ C, int ldc,
    int N, int Kd, int ntiles, int total_tiles)
{
    int wave_id = (int)((blockIdx.x * blockDim.x + threadIdx.x) >> 5);
    if (wave_id >= total_tiles) return;          // wave-uniform exit
    int mg   = wave_id / ntiles;                 // M-group (MT tiles)
    int nt   = wave_id - mg * ntiles;
    int lane = threadIdx.x & 31;
    int l16  = lane & 15;
    int half = lane >> 4;

    int n = nt * 16 + l16;                        // B col for this lane
    int nc = (n < N) ? n : (N - 1);               // clamped (no OOB)
    float nmask = (n < N) ? 1.0f : 0.0f;          // select, no divergence

    const float* Arow[MT];
#pragma unroll
    for (int q = 0; q < MT; ++q)
        Arow[q] = A + (size_t)((mg * MT + q) * 16 + l16) * lda + half * 2;

    const v8f vzero = {0.f, 0.f, 0.f, 0.f, 0.f, 0.f, 0.f, 0.f};
    v8f acc[MT];
#pragma unroll
    for (int q = 0; q < MT; ++q) acc[q] = vzero;

    for (int kk = 0; kk < Kd; kk += 4) {
        v2f b;
        int ka = kk + half * 2;
        if (BT) {
            const float* Bp = B + (size_t)nc * ldb + ka;
            b.x = Bp[0] * nmask;
            b.y = Bp[1] * nmask;
        } else {
            b.x = B[(size_t)ka * ldb + nc] * nmask;
            b.y = B[(size_t)(ka + 1) * ldb + nc] * nmask;
        }
#pragma unroll
        for (int q = 0; q < MT; ++q) {            // MT independent WMMAs / B frag
            v2f a;
            a.x = Arow[q][kk];
            a.y = Arow[q][kk + 1];
            acc[q] = __builtin_amdgcn_wmma_f32_16x16x4_f32(false, a, false, b,
                                                           (short)0, acc[q],
                                                           false, false);
        }
    }

    float bv = bias ? bias[nc] : 0.0f;
#pragma unroll
    for (int q = 0; q < MT; ++q) {
#pragma unroll
        for (int r = 0; r < 8; ++r) {
            int row = (mg * MT + q) * 16 + r + 8 * half;
            int col = nt * 16 + l16;
            if (col < N) {
                size_t off = (size_t)row * ldc + col;
                float v = acc[q][r] + bv;
                if (ACC) C[off] += v; else C[off] = v;
            }
        }
    }
}

// mean over pixels: mean[img][d] = avg_p enc[img][p][d]
__global__ void mean_kernel(const float* __restrict__ enc, float* __restrict__ meanb)
{
    int img = blockIdx.x;
    const float* e = enc + (size_t)img * PPIX * DENC;
    for (int d = threadIdx.x; d < DENC; d += blockDim.x) {
        float s = 0.f;
        for (int p = 0; p < PPIX; ++p) s += e[(size_t)p * DENC + d];
        meanb[img * DENC + d] = s * (1.0f / (float)PPIX);
    }
}

__global__ void init_state_kernel(const float* __restrict__ hinit,
                                  const float* __restrict__ cinit,
                                  float* __restrict__ h, float* __restrict__ c,
                                  float* __restrict__ scores,
                                  int* __restrict__ prev_words,
                                  int* __restrict__ finished,
                                  int* __restrict__ seqs)
{
    int img = blockIdx.x;
    for (int i = threadIdx.x; i < KBEAM * DDIM; i += blockDim.x) {
        int k = i / DDIM, d = i % DDIM;
        h[(size_t)(img * KBEAM + k) * DDIM + d] = hinit[img * DDIM + d];
        c[(size_t)(img * KBEAM + k) * DDIM + d] = cinit[img * DDIM + d];
    }
    for (int i = threadIdx.x; i < KBEAM * SLEN; i += blockDim.x)
        seqs[img * KBEAM * SLEN + i] = STARTW;
    if (threadIdx.x < KBEAM) {
        int r = img * KBEAM + threadIdx.x;
        scores[r] = (threadIdx.x == 0) ? 0.0f : NEGINF;   // step-1 topk trick
        prev_words[r] = STARTW;
        finished[r] = 0;
    }
}

// attention + softmax + awe*gate + x = [emb, awe] assembly (per image block)
__global__ void attention_kernel(const float* __restrict__ att1,
                                 const float* __restrict__ att2,
                                 const float* __restrict__ w_full,
                                 const float* __restrict__ b_full,
                                 const float* __restrict__ fbpre,
                                 const float* __restrict__ enc,
                                 const float* __restrict__ embedding,
                                 const int*   __restrict__ prev_words,
                                 float* __restrict__ x)
{
    __shared__ float s_w[ADIM];
    __shared__ float s_att2[KBEAM][ADIM];
    __shared__ float s_alpha[KBEAM][PPIX];
    int img = blockIdx.x, tid = threadIdx.x, r0 = img * KBEAM;

    for (int i = tid; i < ADIM; i += blockDim.x) s_w[i] = w_full[i];
    for (int i = tid; i < KBEAM * ADIM; i += blockDim.x)
        s_att2[i / ADIM][i % ADIM] = att2[(size_t)(r0 + i / ADIM) * ADIM + (i % ADIM)];
    __syncthreads();

    int wave = tid >> 5, lane = tid & 31, nwaves = blockDim.x >> 5;
    float bf = b_full[0];
    for (int kp = wave; kp < KBEAM * PPIX; kp += nwaves) {
        int k = kp / PPIX, p = kp % PPIX;
        const float* a1 = att1 + ((size_t)img * PPIX + p) * ADIM;
        float s = 0.f;
        for (int a = lane; a < ADIM; a += 32) {
            float t = a1[a] + s_att2[k][a];
            s += fmaxf(t, 0.f) * s_w[a];
        }
        for (int off = 16; off > 0; off >>= 1) s += __shfl_down(s, off, 32);
        if (lane == 0) s_alpha[k][p] = s + bf;
    }
    __syncthreads();

    if (tid < KBEAM) {                         // softmax over 196 (tiny)
        float mx = -1e30f;
        for (int p = 0; p < PPIX; ++p) mx = fmaxf(mx, s_alpha[tid][p]);
        float sum = 0.f;
        for (int p = 0; p < PPIX; ++p) { float v = expf(s_alpha[tid][p] - mx); s_alpha[tid][p] = v; sum += v; }
        float inv = 1.0f / sum;
        for (int p = 0; p < PPIX; ++p) s_alpha[tid][p] *= inv;
    }
    __syncthreads();

    for (int i = tid; i < KBEAM * EDIM; i += blockDim.x) {   // emb part of x
        int k = i / EDIM, j = i % EDIM;
        x[(size_t)(r0 + k) * XDIM + j] = embedding[(size_t)prev_words[r0 + k] * EDIM + j];
    }
    const float* e = enc + (size_t)img * PPIX * DENC;        // awe part (enc read once)
    for (int d = tid; d < DENC; d += blockDim.x) {
        float a0 = 0.f, a1v = 0.f, a2 = 0.f;
        for (int p = 0; p < PPIX; ++p) {
            float v = e[(size_t)p * DENC + d];
            a0  += s_alpha[0][p] * v;
            a1v += s_alpha[1][p] * v;
            a2  += s_alpha[2][p] * v;
        }
        x[(size_t)(r0 + 0) * XDIM + EDIM + d] = sigm(fbpre[(size_t)(r0 + 0) * DENC + d]) * a0;
        x[(size_t)(r0 + 1) * XDIM + EDIM + d] = sigm(fbpre[(size_t)(r0 + 1) * DENC + d]) * a1v;
        x[(size_t)(r0 + 2) * XDIM + EDIM + d] = sigm(fbpre[(size_t)(r0 + 2) * DENC + d]) * a2;
    }
}

__global__ void lstm_kernel(const float* __restrict__ gates,
                            const float* __restrict__ c,
                            float* __restrict__ h_new, float* __restrict__ c_new)
{
    int r = blockIdx.x;
    const float* g = gates + (size_t)r * GDIM;
    for (int d = threadIdx.x; d < DDIM; d += blockDim.x) {
        float ig = sigm(g[d]);
        float fg = sigm(g[DDIM + d]);
        float gg = tanhf(g[2 * DDIM + d]);
        float og = sigm(g[3 * DDIM + d]);
        float cn = fg * c[(size_t)r * DDIM + d] + ig * gg;
        c_new[(size_t)r * DDIM + d] = cn;
        h_new[(size_t)r * DDIM + d] = og * tanhf(cn);
    }
}

__global__ void lse_kernel(const float* __restrict__ logits,
                           float* __restrict__ rowmax, float* __restrict__ rowlse)
{
    __shared__ float red[256];
    int r = blockIdx.x, tid = threadIdx.x;
    const float* L = logits + (size_t)r * VVOC;
    float mx = -1e30f;
    for (int v = tid; v < VVOC; v += blockDim.x) mx = fmaxf(mx, L[v]);
    red[tid] = mx; __syncthreads();
    for (int s = blockDim.x / 2; s > 0; s >>= 1) {
        if (tid < s) red[tid] = fmaxf(red[tid], red[tid + s]);
        __syncthreads();
    }
    mx = red[0]; __syncthreads();
    float sum = 0.f;
    for (int v = tid; v < VVOC; v += blockDim.x) sum += expf(L[v] - mx);
    red[tid] = sum; __syncthreads();
    for (int s = blockDim.x / 2; s > 0; s >>= 1) {
        if (tid < s) red[tid] += red[tid + s];
        __syncthreads();
    }
    if (tid == 0) { rowmax[r] = mx; rowlse[r] = logf(red[0]); }
}

// per-image top-3 over K*V candidates + beam-state reorder (jax tie-break: index)
__global__ void topk_kernel(const float* __restrict__ logits,
                            const float* __restrict__ rowmax,
                            const float* __restrict__ rowlse,
                            float* __restrict__ scores,
                            int* __restrict__ finished,
                            int* __restrict__ prev_words,
                            int* __restrict__ seqs,
                            float* __restrict__ h, float* __restrict__ c,
                            const float* __restrict__ h_new,
                            const float* __restrict__ c_new,
                            int t)
{
    __shared__ float cv[256 * 3];
    __shared__ int   ci[256 * 3];
    __shared__ float s_sc[KBEAM];
    __shared__ int   s_fin[KBEAM];
    __shared__ int   s_prev[KBEAM], s_nxt[KBEAM], s_newf[KBEAM];
    __shared__ float s_news[KBEAM];
    __shared__ int   s_seq[KBEAM][SLEN];

    int img = blockIdx.x, tid = threadIdx.x, r0 = img * KBEAM;
    if (tid < KBEAM) { s_sc[tid] = scores[r0 + tid]; s_fin[tid] = finished[r0 + tid]; }
    for (int i = tid; i < KBEAM * SLEN; i += blockDim.x)
        s_seq[i / SLEN][i % SLEN] = seqs[r0 * SLEN + i];
    __syncthreads();

    float bv0 = -3e9f, bv1 = -3e9f, bv2 = -3e9f;
    int   bi0 = 0x7fffffff, bi1 = 0x7fffffff, bi2 = 0x7fffffff;
    for (int k = 0; k < KBEAM; ++k) {
        float base = s_sc[k];
        float mk = rowmax[r0 + k] + rowlse[r0 + k];
        const float* L = logits + (size_t)(r0 + k) * VVOC;
        bool fin = s_fin[k] != 0;
        for (int v = tid; v < VVOC; v += blockDim.x) {
            float lp = fin ? ((v == ENDW) ? 0.f : NEGINF) : (L[v] - mk);
            float val = base + lp;
            int idx = k * VVOC + v;
            if (val > bv0 || (val == bv0 && idx < bi0)) {
                bv2 = bv1; bi2 = bi1; bv1 = bv0; bi1 = bi0; bv0 = val; bi0 = idx;
            } else if (val > bv1 || (val == bv1 && idx < bi1)) {
                bv2 = bv1; bi2 = bi1; bv1 = val; bi1 = idx;
            } else if (val > bv2 || (val == bv2 && idx < bi2)) {
                bv2 = val; bi2 = idx;
            }
        }
    }
    cv[tid * 3 + 0] = bv0; cv[tid * 3 + 1] = bv1; cv[tid * 3 + 2] = bv2;
    ci[tid * 3 + 0] = bi0; ci[tid * 3 + 1] = bi1; ci[tid * 3 + 2] = bi2;
    __syncthreads();

    if (tid == 0) {
        float g0 = -4e9f, g1 = -4e9f, g2 = -4e9f;
        int   j0 = 0x7fffffff, j1 = 0x7fffffff, j2 = 0x7fffffff;
        for (int i = 0; i < 256 * 3; ++i) {
            float val = cv[i]; int idx = ci[i];
            if (val > g0 || (val == g0 && idx < j0)) { g2 = g1; j2 = j1; g1 = g0; j1 = j0; g0 = val; j0 = idx; }
            else if (val > g1 || (val == g1 && idx < j1)) { g2 = g1; j2 = j1; g1 = val; j1 = idx; }
            else if (val > g2 || (val == g2 && idx < j2)) { g2 = val; j2 = idx; }
        }
        int   pj[3] = { j0 / VVOC, j1 / VVOC, j2 / VVOC };
        int   nj[3] = { j0 % VVOC, j1 % VVOC, j2 % VVOC };
        float gs[3] = { g0, g1, g2 };
        for (int j = 0; j < KBEAM; ++j) {
            s_prev[j] = pj[j]; s_nxt[j] = nj[j]; s_news[j] = gs[j];
            s_newf[j] = (s_fin[pj[j]] != 0) || (nj[j] == ENDW);
        }
    }
    __syncthreads();

    for (int i = tid; i < KBEAM * DDIM; i += blockDim.x) {
        int j = i / DDIM, d = i % DDIM;
        h[(size_t)(r0 + j) * DDIM + d] = h_new[(size_t)(r0 + s_prev[j]) * DDIM + d];
        c[(size_t)(r0 + j) * DDIM + d] = c_new[(size_t)(r0 + s_prev[j]) * DDIM + d];
    }
    for (int i = tid; i < KBEAM * SLEN; i += blockDim.x) {
        int j = i / SLEN, p = i % SLEN;
        seqs[(r0 + j) * SLEN + p] = (p == t) ? s_nxt[j] : s_seq[s_prev[j]][p];
    }
    if (tid < KBEAM) {
        scores[r0 + tid]     = s_news[tid];
        finished[r0 + tid]   = s_newf[tid];
        prev_words[r0 + tid] = s_nxt[tid];
    }
}

__global__ void finalize_kernel(const int* __restrict__ seqs,
                                const float* __restrict__ scores,
                                const int* __restrict__ finished,
                                float* __restrict__ out)
{
    int img = blockIdx.x, tid = threadIdx.x, r0 = img * KBEAM;
    __shared__ int bestk;
    if (tid == 0) {
        bool any = finished[r0] || finished[r0 + 1] || finished[r0 + 2];
        float best = -1e30f; int bk = 0;
        for (int k = 0; k < KBEAM; ++k) {
            float e = scores[r0 + k];
            if (any && !finished[r0 + k]) e = NEGINF;
            if (e > best) { best = e; bk = k; }     // first-max tie-break
        }
        bestk = bk;
    }
    __syncthreads();
    for (int j = tid; j < SLEN; j += blockDim.x)
        out[img * SLEN + j] = (float)seqs[(r0 + bestk) * SLEN + j];
    for (int i = tid; i < KBEAM * SLEN; i += blockDim.x)
        out[BIMG * SLEN + r0 * SLEN + i] = (float)seqs[r0 * SLEN + i];
    if (tid < KBEAM)
        out[BIMG * SLEN + BIMG * KBEAM * SLEN + r0 + tid] = scores[r0 + tid];
}

// ---------------------------------------------------------------------------
extern "C" void kernel_launch(void* const* d_in, const int* in_sizes, int n_in,
                              void* d_out, int out_size, void* d_ws, size_t ws_size,
                              hipStream_t stream)
{
    (void)in_sizes; (void)n_in; (void)out_size; (void)ws_size;

    const float* enc       = (const float*)d_in[0];
    const float* embedding = (const float*)d_in[1];
    const float* W_enc_att = (const float*)d_in[2];
    const float* b_enc_att = (const float*)d_in[3];
    const float* W_dec_att = (const float*)d_in[4];
    const float* b_dec_att = (const float*)d_in[5];
    const float* w_full    = (const float*)d_in[6];
    const float* b_full    = (const float*)d_in[7];
    const float* W_init_h  = (const float*)d_in[8];
    const float* b_init_h  = (const float*)d_in[9];
    const float* W_init_c  = (const float*)d_in[10];
    const float* b_init_c  = (const float*)d_in[11];
    const float* W_fb      = (const float*)d_in[12];
    const float* b_fb      = (const float*)d_in[13];
    const float* W_ih      = (const float*)d_in[14];  // [2048, 2560] (torch)
    const float* b_ih      = (const float*)d_in[15];
    const float* W_hh      = (const float*)d_in[16];  // [2048, 512]  (torch)
    const float* b_hh      = (const float*)d_in[17];
    const float* W_fc      = (const float*)d_in[18];  // [512, 9490]
    const float* b_fc      = (const float*)d_in[19];

    float* ws = (float*)d_ws;
    size_t off = 0;
    auto alloc = [&](size_t n) { float* p = ws + off; off += n; return p; };
    float* att1   = alloc((size_t)BIMG * PPIX * ADIM);   // 1.6M floats
    float* meanb  = alloc((size_t)BIMG * DENC);
    float* hinit  = alloc((size_t)BIMG * DDIM);
    float* cinit  = alloc((size_t)BIMG * DDIM);
    float* h      = alloc((size_t)NROW * DDIM);
    float* c      = alloc((size_t)NROW * DDIM);
    float* h_new  = alloc((size_t)NROW * DDIM);
    float* c_new  = alloc((size_t)NROW * DDIM);
    float* att2   = alloc((size_t)NROW * ADIM);
    float* fbpre  = alloc((size_t)NROW * DENC);
    float* xbuf   = alloc((size_t)NROW * XDIM);
    float* gates  = alloc((size_t)NROW * GDIM);
    float* logits = alloc((size_t)NROW * VVOC);
    float* rowmax = alloc(NROW);
    float* rowlse = alloc(NROW);
    float* scores = alloc(NROW);
    int* prev_words = (int*)alloc(NROW);
    int* finished   = (int*)alloc(NROW);
    int* seqs       = (int*)alloc((size_t)NROW * SLEN);

    auto grid_for = [](int tiles) { return (tiles + 7) / 8; };  // 8 waves/block

    // ---- prep --------------------------------------------------------------
    mean_kernel<<<BIMG, 256, 0, stream>>>(enc, meanb);
    {   // att1 = enc(flat [3136,2048]) @ W_enc_att + b   (hoisted, constant)
        int nt = ADIM / 16, tiles = (BIMG * PPIX / 16 / 2) * nt;  // MT=2
        wmma_gemm_kernel<2, false, false><<<grid_for(tiles), 256, 0, stream>>>(
            enc, DENC, W_enc_att, ADIM, b_enc_att, att1, ADIM, ADIM, DENC, nt, tiles);
    }
    {   // h0,c0 = mean @ W_init_{h,c} + b   (M=16 -> one M tile)
        int nt = DDIM / 16, tiles = nt;
        wmma_gemm_kernel<1, false, false><<<grid_for(tiles), 256, 0, stream>>>(
            meanb, DENC, W_init_h, DDIM, b_init_h, hinit, DDIM, DDIM, DENC, nt, tiles);
        wmma_gemm_kernel<1, false, false><<<grid_for(tiles), 256, 0, stream>>>(
            meanb, DENC, W_init_c, DDIM, b_init_c, cinit, DDIM, DDIM, DENC, nt, tiles);
    }
    init_state_kernel<<<BIMG, 256, 0, stream>>>(hinit, cinit, h, c, scores,
                                                prev_words, finished, seqs);

    // ---- 50 sequential decode steps ---------------------------------------
    for (int t = 1; t <= 50; ++t) {
        {   // att2 = h @ W_dec_att + b     [48,512]x[512,512]  (tiny B: MT=1)
            int nt = ADIM / 16, tiles = 3 * nt;
            wmma_gemm_kernel<1, false, false><<<grid_for(tiles), 256, 0, stream>>>(
                h, DDIM, W_dec_att, ADIM, b_dec_att, att2, ADIM, ADIM, DDIM, nt, tiles);
        }
        {   // gate pre-act = h @ W_fb + b   [48,512]x[512,2048]  (MT=3)
            int nt = DENC / 16, tiles = nt;
            wmma_gemm_kernel<3, false, false><<<grid_for(tiles), 256, 0, stream>>>(
                h, DDIM, W_fb, DENC, b_fb, fbpre, DENC, DENC, DDIM, nt, tiles);
        }
        attention_kernel<<<BIMG, 256, 0, stream>>>(att1, att2, w_full, b_full,
                                                   fbpre, enc, embedding, prev_words, xbuf);
        {   // gates = x @ W_ih^T + b_ih ; += h @ W_hh^T + b_hh   (MT=3)
            int nt = GDIM / 16, tiles = nt;
            wmma_gemm_kernel<3, true, false><<<grid_for(tiles), 256, 0, stream>>>(
                xbuf, XDIM, W_ih, XDIM, b_ih, gates, GDIM, GDIM, XDIM, nt, tiles);
            wmma_gemm_kernel<3, true, true><<<grid_for(tiles), 256, 0, stream>>>(
                h, DDIM, W_hh, DDIM, b_hh, gates, GDIM, GDIM, DDIM, nt, tiles);
        }
        lstm_kernel<<<NROW, 256, 0, stream>>>(gates, c, h_new, c_new);
        {   // logits = h_new @ W_fc + b_fc   [48,512]x[512,9490]  (MT=3)
            int nt = (VVOC + 15) / 16, tiles = nt;
            wmma_gemm_kernel<3, false, false><<<grid_for(tiles), 256, 0, stream>>>(
                h_new, DDIM, W_fc, VVOC, b_fc, logits, VVOC, VVOC, DDIM, nt, tiles);
        }
        lse_kernel<<<NROW, 256, 0, stream>>>(logits, rowmax, rowlse);
        topk_kernel<<<BIMG, 256, 0, stream>>>(logits, rowmax, rowlse, scores,
                                              finished, prev_words, seqs,
                                              h, c, h_new, c_new, t);
    }

    finalize_kernel<<<BIMG, 64, 0, stream>>>(seqs, scores, finished, (float*)d_out);
}